// CovLoss_89335319757215
// MI455X (gfx1250) — compile-verified
//
#include <hip/hip_runtime.h>
#include <hip/hip_bf16.h>

// ---------------------------------------------------------------------------
// CDNA5 wave32 full-lane f32 sum using V_WMMA_F32_16X16X4_F32.
//   A (16x4): lane L<16  holds A(L,0)=x, A(L,1)=0
//             lane L>=16 holds A(L-16,2)=x, A(L-16,3)=0
//   B = ones  ->  D(m,n) = x[m] + x[m+16]  (row sums, replicated over n)
//   D layout: lane l, reg r:  l<16 -> M=r ; l>=16 -> M=8+r
//   sum of 8 D regs = half-row-range total; shfl_xor(16) adds the other half.
// Result: every lane returns the sum of x over all 32 lanes (pure f32 path).
// EXEC must be all ones at the call site (call uniformly from all threads).
// ---------------------------------------------------------------------------
typedef __attribute__((ext_vector_type(2))) float v2f;
typedef __attribute__((ext_vector_type(8))) float v8f;

__device__ __forceinline__ float wave_sum32(float x) {
    v2f a; a[0] = x;    a[1] = 0.0f;
    v2f b; b[0] = 1.0f; b[1] = 1.0f;
    v8f c = {};
    v8f d = __builtin_amdgcn_wmma_f32_16x16x4_f32(
        /*neg_a=*/false, a, /*neg_b=*/false, b,
        /*c_mod=*/(short)0, c, /*reuse_a=*/false, /*reuse_b=*/false);
    float s = 0.0f;
#pragma unroll
    for (int i = 0; i < 8; ++i) s += d[i];
    s += __shfl_xor(s, 16);
    return s;
}

// Workspace layout (floats):
//  [0]        bce_sum (global)
//  [1..32]    per-sample intersection
//  [33..64]   per-sample mask sum
//  [65..96]   per-sample label sum
//  [97..416]  10 correlation moment arrays x 32 samples:
//             Sv,Sh,Sl,Svv,Shh,Sll,Svh,Svl,Shl,Svhl
#define WS_BCE   0
#define WS_INTER 1
#define WS_MSUM  33
#define WS_LSUM  65
#define WS_CORR  97
#define WS_FLOATS 417

#define NSAMP 32
#define HW    512
#define SELEM (HW * HW)          // 262144 per sample
#define PDIM  128
#define PELEM (PDIM * PDIM)      // 16384 pooled positions per sample

// ---------------------------------------------------------------------------
// Pass 1: fused BCE + Dice partial sums. grid = (32 chunks, 32 samples),
// 256 threads; each block streams 8192 elements (2048 float4 pairs).
// ---------------------------------------------------------------------------
__global__ void bce_dice_kernel(const float* __restrict__ logits,
                                const float* __restrict__ labels,
                                float* __restrict__ ws) {
    const int sample   = blockIdx.y;
    const int chunk_f4 = (SELEM / 4) / 32;  // 2048 float4 per block
    const long base4   = (long)sample * (SELEM / 4) + (long)blockIdx.x * chunk_f4;
    const float4* lp = (const float4*)logits + base4;
    const float4* yp = (const float4*)labels + base4;

    float bce = 0.0f, inter = 0.0f, msum = 0.0f, lsum = 0.0f;

#define ACC1(px, yx)                                            \
    {                                                           \
        bce += (yx) * __logf(px) + (1.0f - (yx)) * __logf(1.0f - (px)); \
        float m_ = ((px) > 0.4f) ? 1.0f : 0.0f;                 \
        inter += m_ * (yx);                                     \
        msum  += m_;                                            \
        lsum  += (yx);                                          \
    }

    for (int i = threadIdx.x; i < chunk_f4; i += blockDim.x) {
        float4 p = lp[i];
        float4 y = yp[i];
        ACC1(p.x, y.x) ACC1(p.y, y.y) ACC1(p.z, y.z) ACC1(p.w, y.w)
    }
#undef ACC1

    bce   = wave_sum32(bce);
    inter = wave_sum32(inter);
    msum  = wave_sum32(msum);
    lsum  = wave_sum32(lsum);

    if ((threadIdx.x & 31) == 0) {
        atomicAdd(&ws[WS_BCE], bce);
        atomicAdd(&ws[WS_INTER + sample], inter);
        atomicAdd(&ws[WS_MSUM  + sample], msum);
        atomicAdd(&ws[WS_LSUM  + sample], lsum);
    }
}

// ---------------------------------------------------------------------------
// Pass 2: correlation moment sums. grid = (16 chunks, 32 samples), 256 thr.
// Each pooled position reads v, h, and 4 float4 rows of labels (4x4 mean).
// ---------------------------------------------------------------------------
__global__ void corr_kernel(const float* __restrict__ labels,
                            const float* __restrict__ v_att,
                            const float* __restrict__ h_att,
                            float* __restrict__ ws) {
    const int sample    = blockIdx.y;
    const int per_block = PELEM / 16;  // 1024 positions
    const int start     = blockIdx.x * per_block;
    const float4* lab4 = (const float4*)labels + (long)sample * (SELEM / 4);
    const float*  vp   = v_att + (long)sample * PELEM;
    const float*  hp   = h_att + (long)sample * PELEM;

    float Sv = 0, Sh = 0, Sl = 0, Svv = 0, Shh = 0, Sll = 0;
    float Svh = 0, Svl = 0, Shl = 0, Svhl = 0;

    for (int i = start + (int)threadIdx.x; i < start + per_block; i += blockDim.x) {
        const int py = i >> 7;
        const int px = i & 127;
        float l = 0.0f;
#pragma unroll
        for (int r = 0; r < 4; ++r) {
            float4 t = lab4[(4 * py + r) * (HW / 4) + px];
            l += t.x + t.y + t.z + t.w;
        }
        l *= 0.0625f;  // mean of 4x4 block
        const float v = vp[i];
        const float h = hp[i];
        Sv += v;       Sh += h;       Sl += l;
        Svv += v * v;  Shh += h * h;  Sll += l * l;
        Svh += v * h;  Svl += v * l;  Shl += h * l;
        Svhl += v * h * l;
    }

    float sums[10] = {Sv, Sh, Sl, Svv, Shh, Sll, Svh, Svl, Shl, Svhl};
#pragma unroll
    for (int k = 0; k < 10; ++k) sums[k] = wave_sum32(sums[k]);

    if ((threadIdx.x & 31) == 0) {
#pragma unroll
        for (int k = 0; k < 10; ++k)
            atomicAdd(&ws[WS_CORR + k * NSAMP + sample], sums[k]);
    }
}

// ---------------------------------------------------------------------------
// Pass 3: finalize. One wave; lane s handles sample s, then WMMA-reduces.
// Centered sums recovered algebraically:
//   sum(vc*hc*lc) = Svhl - ml*Svh - mh*Svl - mv*Shl + 2*K*mv*mh*ml
//   sum(vc*vc)    = Svv - K*mv^2   (etc.)
// ---------------------------------------------------------------------------
__global__ void finalize_kernel(const float* __restrict__ ws,
                                float* __restrict__ out) {
    const int s   = threadIdx.x;  // 0..31 == sample id
    const float K = (float)PELEM;

    const float inter = ws[WS_INTER + s];
    const float msum  = ws[WS_MSUM  + s];
    const float lsum  = ws[WS_LSUM  + s];
    const float dice  = 2.0f * (inter + 1.0f) / (msum + lsum + 1.0f);

    const float* C = ws + WS_CORR;
    const float Sv   = C[0 * NSAMP + s], Sh  = C[1 * NSAMP + s], Sl  = C[2 * NSAMP + s];
    const float Svv  = C[3 * NSAMP + s], Shh = C[4 * NSAMP + s], Sll = C[5 * NSAMP + s];
    const float Svh  = C[6 * NSAMP + s], Svl = C[7 * NSAMP + s], Shl = C[8 * NSAMP + s];
    const float Svhl = C[9 * NSAMP + s];

    const float mv = Sv / K, mh = Sh / K, ml = Sl / K;
    const float num = Svhl - ml * Svh - mh * Svl - mv * Shl + 2.0f * K * mv * mh * ml;
    const float den = sqrtf((Svv - K * mv * mv) * (Shh - K * mh * mh) * (Sll - K * ml * ml));
    const float cor = num / den;

    const float dsum = wave_sum32(dice);
    const float csum = wave_sum32(cor);

    if (s == 0) {
        const float total    = (float)NSAMP * (float)SELEM;  // 8388608
        const float bceloss  = -ws[WS_BCE] / total;
        const float diceloss = 1.0f - dsum / (float)NSAMP;
        const float corloss  = -csum / (float)NSAMP;
        out[0] = 0.2f * bceloss + 0.3f * diceloss + 0.5f * corloss;
    }
}

extern "C" void kernel_launch(void* const* d_in, const int* in_sizes, int n_in,
                              void* d_out, int out_size, void* d_ws, size_t ws_size,
                              hipStream_t stream) {
    const float* logits = (const float*)d_in[0];
    const float* labels = (const float*)d_in[1];
    const float* v_att  = (const float*)d_in[2];
    const float* h_att  = (const float*)d_in[3];
    float* out = (float*)d_out;
    float* ws  = (float*)d_ws;

    // Accumulators must start at zero every call (graph-capture-safe memset).
    hipMemsetAsync(d_ws, 0, WS_FLOATS * sizeof(float), stream);

    bce_dice_kernel<<<dim3(32, NSAMP), 256, 0, stream>>>(logits, labels, ws);
    corr_kernel<<<dim3(16, NSAMP), 256, 0, stream>>>(labels, v_att, h_att, ws);
    finalize_kernel<<<1, 32, 0, stream>>>(ws, out);
}